// GFSQ_51256139710669
// MI455X (gfx1250) — compile-verified
//
#include <hip/hip_runtime.h>
#include <hip/hip_bf16.h>

// GFSQ on MI455X (gfx1250). Memory-bound: x read 128MiB + feat write 128MiB
// dominate (~11.5us floor at 23.3TB/s). Both skinny GEMMs use
// V_WMMA_F32_16X16X4_F32 (exact K=4 fit for project_out; K-step for project_in).

typedef float v2f __attribute__((ext_vector_type(2)));
typedef float v8f __attribute__((ext_vector_type(8)));

static __device__ __forceinline__ v8f wmma_f32_k4(v2f a, v2f b, v8f c) {
    // 8-arg pattern: (neg_a, A, neg_b, B, c_mod, C, reuse_a, reuse_b)
    return __builtin_amdgcn_wmma_f32_16x16x4_f32(false, a, false, b, (short)0, c,
                                                 false, false);
}

#define TT 2048      // T
#define DIMX 1024    // DIM
#define NBINS 625
#define NWd 16384    // 1024 x 16 padded project_in weight
#define NWo 8192     // 1024 x 8  dense project_out weight (dim-major)
#define NC  2500     // 4 x 625 histogram

// ---- setup: build dense padded weights, zero histogram + "zero" output ----
__global__ void k_setup(const float* __restrict__ w_in,  // (2,4,512)
                        const float* __restrict__ w_out, // (2,512,4)
                        float* __restrict__ Wd, float* __restrict__ Wo,
                        float* __restrict__ counts, float* __restrict__ out_zero) {
    int i = blockIdx.x * blockDim.x + threadIdx.x;
    int stride = gridDim.x * blockDim.x;
    for (; i < NWd + NWo + NC + 4; i += stride) {
        if (i < NWd) {                      // Wd[k][n], n = g*4+c, block-diagonal
            int k = i >> 4, n = i & 15;
            float v = 0.0f;
            if (n < 8) {
                int g = n >> 2, c = n & 3;
                if ((k >> 9) == g) v = w_in[(g * 4 + c) * 512 + (k & 511)];
            }
            Wd[i] = v;
        } else if (i < NWd + NWo) {         // Wo[m][n]: m=dim, n = g*4+c
            int j = i - NWd;
            int m = j >> 3, n = j & 7;
            int g = n >> 2, c = n & 3;
            Wo[j] = ((m >> 9) == g) ? w_out[m * 4 + c] : 0.0f;
        } else if (i < NWd + NWo + NC) {
            counts[i - NWd - NWo] = 0.0f;
        } else {
            out_zero[i - NWd - NWo - NC] = 0.0f;
        }
    }
}

// ---- project_in: z[token][8] = x-tile(16 tok x 1024) * Wd(1024x16) + b_in ----
// One wave per 16-token tile; 256 chained f32 16x16x4 WMMAs over K=1024.
__global__ __launch_bounds__(256) void k_project_in(
    const float* __restrict__ x, const float* __restrict__ Wd,
    const float* __restrict__ b_in, float* __restrict__ z_ws) {
    int wave = blockIdx.x * 8 + (threadIdx.x >> 5);
    int lane = threadIdx.x & 31;
    int half = lane >> 4, lm = lane & 15;
    int b = wave >> 7;                 // 128 tiles of 16 tokens per batch
    int t0 = (wave & 127) << 4;

    // A (16 tok x 4 dim): lane halves hold K pairs {0,1} / {2,3}
    const float* xA = x + ((size_t)b * DIMX + 2 * half) * TT + t0 + lm;
    const float* wB = Wd + (2 * half) * 16 + lm;   // B (4 x 16): lane = N column

    v8f acc = {};
#pragma unroll 4
    for (int k = 0; k < DIMX; k += 4) {
        v2f a, w;
        a.x = xA[0];
        a.y = xA[TT];
        w.x = wB[0];
        w.y = wB[16];
        xA += 4 * (size_t)TT;
        wB += 64;
        acc = wmma_f32_k4(a, w, acc);
    }
    // D: VGPR r -> token t0+r+8*half, lane -> column; keep 8 real columns
    if (lm < 8) {
        float bias = b_in[lm];
        size_t base = ((size_t)b * TT + t0 + 8 * half) * 8 + lm;
#pragma unroll
        for (int r = 0; r < 8; ++r)
            z_ws[base + (size_t)r * 8] = acc[r] + bias;
    }
}

// ---- residual FSQ quantize (R=2, levels=5): q, indices, histogram ----
__global__ __launch_bounds__(256) void k_quantize(
    const float* __restrict__ z_ws, float* __restrict__ q_ws,
    float* __restrict__ counts, float* __restrict__ out_ind) {
    int tid = blockIdx.x * 256 + threadIdx.x;   // 0..32767 token id
    int b = tid >> 11, t = tid & 2047;
    float4 zlo = *(const float4*)(z_ws + (size_t)tid * 8);
    float4 zhi = *(const float4*)(z_ws + (size_t)tid * 8 + 4);
    float z8[8] = {zlo.x, zlo.y, zlo.z, zlo.w, zhi.x, zhi.y, zhi.z, zhi.w};

    const float inv_s[2] = {1.0f, 4.0f};   // 1/scale per residual round
    const float sc[2]    = {1.0f, 0.25f};
    const int basis[4]   = {1, 5, 25, 125};
#pragma unroll
    for (int g = 0; g < 2; ++g) {
        float resid[4], qsum[4] = {0.f, 0.f, 0.f, 0.f};
#pragma unroll
        for (int c = 0; c < 4; ++c) resid[c] = z8[g * 4 + c];
#pragma unroll
        for (int r = 0; r < 2; ++r) {
            int id = 0;
#pragma unroll
            for (int c = 0; c < 4; ++c) {
                // levels=5 (odd): shift=0, half_l=2.002, half_width=2
                float bounded = tanhf(resid[c] * inv_s[r]) * 2.002f;
                float rounded = rintf(bounded);          // RNE like jnp.round
                float q = rounded * 0.5f * sc[r];
                resid[c] -= q;
                qsum[c] += q;
                id += ((int)rounded + 2) * basis[c];
            }
            int j = g * 2 + r;
            out_ind[((size_t)b * 4 + j) * TT + t] = (float)id;
            atomicAdd(counts + j * NBINS + id, 1.0f);
        }
#pragma unroll
        for (int c = 0; c < 4; ++c) q_ws[(size_t)tid * 8 + g * 4 + c] = qsum[c];
    }
}

// ---- project_out, transposed: feat_t[dim][t] = Wo(1024x8) * q^T(8 x tok) ----
// A = Wo tile (M=dim), B = q^T tile (N=token) -> coalesced stores along T.
__global__ __launch_bounds__(256) void k_project_out(
    const float* __restrict__ q_ws, const float* __restrict__ Wo,
    const float* __restrict__ b_out, float* __restrict__ out_feat) {
    int wave = blockIdx.x * 8 + (threadIdx.x >> 5);
    int lane = threadIdx.x & 31;
    int half = lane >> 4, lm = lane & 15;
    int b = wave >> 7;
    int t0 = (wave & 127) << 4;
    int kk = 2 * half;

    // B (K x 16 tokens): lane = token, lane-halves hold K pairs; reused 64x
    const float* qrow = q_ws + ((size_t)b * TT + t0 + lm) * 8;
    v2f B0 = *(const v2f*)(qrow + kk);        // K = {0,1}/{2,3}
    v2f B1 = *(const v2f*)(qrow + 4 + kk);    // K = {4,5}/{6,7}

    for (int n0 = 0; n0 < DIMX; n0 += 16) {
        const float* wrow = Wo + (size_t)(n0 + lm) * 8;   // A: lane = dim row
        v2f A0 = *(const v2f*)(wrow + kk);
        v2f A1 = *(const v2f*)(wrow + 4 + kk);
        v8f acc = {};
        acc = wmma_f32_k4(A0, B0, acc);
        acc = wmma_f32_k4(A1, B1, acc);
#pragma unroll
        for (int r = 0; r < 8; ++r) {
            int dim = n0 + r + 8 * half;
            out_feat[((size_t)b * DIMX + dim) * TT + t0 + lm] = acc[r] + b_out[dim];
        }
    }
}

// ---- perplexity from 4x625 histogram; one wave per codebook row ----
__global__ __launch_bounds__(128) void k_perplexity(
    const float* __restrict__ counts, float* __restrict__ out_perp) {
    int j = threadIdx.x >> 5;
    int lane = threadIdx.x & 31;
    const float BT = 32768.0f;
    float tot = 0.0f;
    for (int i = lane; i < NBINS; i += 32) tot += counts[j * NBINS + i];
#pragma unroll
    for (int m = 16; m >= 1; m >>= 1) tot += __shfl_xor(tot, m, 32);
    float denom = tot / BT + 1e-5f;
    float h = 0.0f;
    for (int i = lane; i < NBINS; i += 32) {
        float e = (counts[j * NBINS + i] / BT) / denom;
        h += e * logf(e + 1e-5f);
    }
#pragma unroll
    for (int m = 16; m >= 1; m >>= 1) h += __shfl_xor(h, m, 32);
    if (lane == 0) out_perp[j] = expf(-h);
}

extern "C" void kernel_launch(void* const* d_in, const int* in_sizes, int n_in,
                              void* d_out, int out_size, void* d_ws, size_t ws_size,
                              hipStream_t stream) {
    const float* x     = (const float*)d_in[0];   // (16,1024,2048)
    const float* w_in  = (const float*)d_in[1];   // (2,4,512)
    const float* b_in  = (const float*)d_in[2];   // (2,4)
    const float* w_out = (const float*)d_in[3];   // (2,512,4)
    const float* b_out = (const float*)d_in[4];   // (2,512) == bias per dim

    float* out      = (float*)d_out;
    float* out_zero = out;                                  // 4
    float* out_feat = out + 4;                              // 16*1024*2048
    float* out_perp = out + 4 + (size_t)16 * 1024 * 2048;   // 4
    float* out_ind  = out_perp + 4;                         // 16*4*2048

    // workspace layout (floats): ~2.2 MiB total
    float* ws     = (float*)d_ws;
    float* Wd     = ws;                    // 16384
    float* Wo     = ws + NWd;              // 8192
    float* counts = ws + NWd + NWo;        // 2500
    float* z_ws   = ws + 27136;            // 32768*8 (16B aligned)
    float* q_ws   = z_ws + (size_t)32768 * 8;

    k_setup<<<64, 256, 0, stream>>>(w_in, w_out, Wd, Wo, counts, out_zero);
    k_project_in<<<256, 256, 0, stream>>>(x, Wd, b_in, z_ws);
    k_quantize<<<128, 256, 0, stream>>>(z_ws, q_ws, counts, out_ind);
    k_project_out<<<256, 256, 0, stream>>>(q_ws, Wo, b_out, out_feat);
    k_perplexity<<<1, 128, 0, stream>>>(counts, out_perp);
}